// Encoder_11269994185007
// MI455X (gfx1250) — compile-verified
//
#include <hip/hip_runtime.h>

// ---------------------------------------------------------------------------
// RGCN layer, fused:  out = sum_s A[s] @ Y[s],
//   Y[s][n][e] = sum_f features[n][f] * V2[s*32+f][e]
//   V2[c][e]   = sum_b W_comp[c%4][b] * W[b][c/4][e]   (faithful to the
//                reference's mismatched reshape: supports_ col c pairs with
//                V row c, where V rows are f-major).
// Dominant cost: streaming A (1.07 GB) once -> HBM-bound (~46us @ 23.3TB/s).
// fp32 WMMA 16x16x4 keeps reference precision; compute is far below roofline.
// A loads: global_load_b64 th:TH_LOAD_NT via explicit addrspace(1) pointers
// (NOT flat) so only LOADcnt is used and the 1.07 GB stream does not evict
// the L2-resident Y (heavily reused B operand) from the 192 MB L2.
// ---------------------------------------------------------------------------

#define NN   8192
#define FDIM 32
#define EDIM 32
#define SUP  4
#define NB   2

#define NCHUNK 4                  // m-dimension split for parallelism
#define CHUNK  (NN / NCHUNK)      // 2048
#define NSUB   4                  // 16-row subtiles per wave -> 64 rows/wave
#define ROWS_PER_WG (NSUB * 16)   // 64
#define NRG    (NN / ROWS_PER_WG) // 128 row groups

typedef __attribute__((ext_vector_type(2))) float v2f;
typedef __attribute__((ext_vector_type(8))) float v8f;

#define GLOBAL_AS __attribute__((address_space(1)))
typedef const GLOBAL_AS float* gfloatp;
typedef const GLOBAL_AS v2f*   gv2fp;

// ---------------------------------------------------------------------------
// Kernel 1: Y[s][n][e] = sum_f features[n][f] * V2[s*32+f][e]
//   c = s*32 + f  ->  s' = c%4 = f&3,  f' = c/4 = s*8 + (f>>2)
//   V2[c][e] = W_comp[s'][0]*W[0][f'][e] + W_comp[s'][1]*W[1][f'][e]
// 67 MFLOP total; negligible vs. main pass. W/W_comp/features are L2-hot.
// ---------------------------------------------------------------------------
__global__ void rgcn_embed(const float* __restrict__ features,
                           const float* __restrict__ W,
                           const float* __restrict__ Wc,
                           float* __restrict__ Y) {
    int g = blockIdx.x * blockDim.x + threadIdx.x;   // 0 .. S*N*E-1
    if (g >= SUP * NN * EDIM) return;
    int e = g & (EDIM - 1);
    int s = (g >> 5) & (SUP - 1);
    int n = g >> 7;

    float acc = 0.0f;
#pragma unroll
    for (int f = 0; f < FDIM; ++f) {
        int sp = f & 3;                 // c % 4
        int fp = s * 8 + (f >> 2);      // c / 4
        float v = Wc[sp * NB + 0] * W[fp * EDIM + e]
                + Wc[sp * NB + 1] * W[FDIM * EDIM + fp * EDIM + e];
        acc = fmaf(features[n * FDIM + f], v, acc);
    }
    Y[(size_t)s * NN * EDIM + (size_t)n * EDIM + e] = acc;
}

// ---------------------------------------------------------------------------
// Kernel 2: main pass. One wave per workgroup (wave32, EXEC all-ones).
//   block (rg, s, c): rows [rg*64, rg*64+64), relation s, m in [c*2048,...).
//   Per K-step: 4x global_load_b64 NT (A, ISA 16x4 f32 layout), 4x b32 (B
//   from L2-resident Y, 4x16 layout), 8x v_wmma_f32_16x16x4_f32.
//   All loop pointers are strength-reduced (pointer += const) so the hot
//   loop is loads + wmma + pointer increments only.
//   Partials per (s,chunk) -> deterministic fixed-order reduction later.
// ---------------------------------------------------------------------------
__global__ __launch_bounds__(32)
void rgcn_main(const float* __restrict__ A,
               const float* __restrict__ Y,
               float* __restrict__ partial) {
    const int lane  = threadIdx.x;          // 0..31
    const int rg    = blockIdx.x;           // 0..127
    const int s     = blockIdx.y;           // 0..3
    const int c     = blockIdx.z;           // 0..3

    const int laneM  = lane & 15;           // M (A) / N (B,C) index
    const int laneK2 = (lane >> 4) << 1;    // K-pair base: 0 or 2
    const int rowBase = rg * ROWS_PER_WG;
    const int mBase   = c * CHUNK;

    // B pointer: advanced by 4*EDIM floats per K-step (strength-reduced).
    const float* bp = Y + (size_t)s * NN * EDIM
                        + (size_t)(mBase + laneK2) * EDIM + laneM;

    // Per-subtile A row base pointers in explicit GLOBAL address space so
    // nontemporal loads lower to global_load_b64 (LOADcnt only), not flat.
    gfloatp ap[NSUB];
#pragma unroll
    for (int t = 0; t < NSUB; ++t) {
        const float* p = A + (size_t)s * NN * NN
                           + (size_t)(rowBase + t * 16 + laneM) * NN
                           + (size_t)(mBase + laneK2);
        ap[t] = (gfloatp)(unsigned long long)(const void*)p;
    }

    v8f acc[NSUB][2];
#pragma unroll
    for (int t = 0; t < NSUB; ++t)
#pragma unroll
        for (int h = 0; h < 2; ++h)
            acc[t][h] = (v8f)0.0f;

    for (int m = 0; m < CHUNK; m += 4) {
        // B tile: rows (m+kb, m+kb+1), cols h*16 + laneM; 4x b32, one base.
        v2f b0, b1;
        b0.x = bp[0];         // K=kb,   N=laneM        (e-half 0)
        b1.x = bp[16];        // K=kb,   N=laneM+16     (e-half 1)
        b0.y = bp[EDIM];      // K=kb+1, N=laneM
        b1.y = bp[EDIM + 16]; // K=kb+1, N=laneM+16
        bp += 4 * EDIM;

#pragma unroll
        for (int t = 0; t < NSUB; ++t) {
            // A is streamed once: non-temporal so it doesn't evict Y from L2.
            v2f a = __builtin_nontemporal_load((gv2fp)ap[t]);
            ap[t] += 4;
            acc[t][0] = __builtin_amdgcn_wmma_f32_16x16x4_f32(
                false, a, false, b0, (short)0, acc[t][0], false, false);
            acc[t][1] = __builtin_amdgcn_wmma_f32_16x16x4_f32(
                false, a, false, b1, (short)0, acc[t][1], false, false);
        }
    }

    // Store C/D: VGPR r -> M = r (lanes 0-15) / r+8 (lanes 16-31), N = laneM
    float* __restrict__ P = partial + (size_t)(s * NCHUNK + c) * NN * EDIM;
    const int rowHi = (lane >> 4) * 8;
#pragma unroll
    for (int t = 0; t < NSUB; ++t) {
#pragma unroll
        for (int h = 0; h < 2; ++h) {
#pragma unroll
            for (int r = 0; r < 8; ++r) {
                int row = rowBase + t * 16 + rowHi + r;
                int col = h * 16 + laneM;
                P[(size_t)row * EDIM + col] = acc[t][h][r];
            }
        }
    }
}

// ---------------------------------------------------------------------------
// Kernel 3: out[g] = sum_{p=0..15} partial[p][g], fixed order (deterministic).
// ---------------------------------------------------------------------------
__global__ void rgcn_reduce(const float* __restrict__ partial,
                            float* __restrict__ out) {
    int g = blockIdx.x * blockDim.x + threadIdx.x;
    if (g >= NN * EDIM) return;
    float acc = 0.0f;
#pragma unroll
    for (int p = 0; p < SUP * NCHUNK; ++p)
        acc += partial[(size_t)p * NN * EDIM + g];
    out[g] = acc;
}

// ---------------------------------------------------------------------------
// Launch. Inputs (setup_inputs order): features, A, W, W_comp. Output: [N, E]
// fp32. Workspace: Y (4 MB) + partials (16 MB) = 20 MB, fully overwritten
// every call (no reliance on poisoned state; no atomics; deterministic).
// ---------------------------------------------------------------------------
extern "C" void kernel_launch(void* const* d_in, const int* in_sizes, int n_in,
                              void* d_out, int out_size, void* d_ws, size_t ws_size,
                              hipStream_t stream) {
    const float* features = (const float*)d_in[0];
    const float* A        = (const float*)d_in[1];
    const float* W        = (const float*)d_in[2];
    const float* Wc       = (const float*)d_in[3];
    float* out = (float*)d_out;

    float* Y       = (float*)d_ws;
    float* partial = (float*)((char*)d_ws + (size_t)SUP * NN * EDIM * sizeof(float));

    // Stage 1: Y[s] = features @ V2_s   (tiny)
    {
        int total = SUP * NN * EDIM;
        rgcn_embed<<<(total + 255) / 256, 256, 0, stream>>>(features, W, Wc, Y);
    }
    // Stage 2: partial[s,c] = A[s][:, chunk_c] @ Y[s][chunk_c]   (WMMA, HBM-bound)
    {
        dim3 grid(NRG, SUP, NCHUNK);
        rgcn_main<<<grid, 32, 0, stream>>>(A, Y, partial);
    }
    // Stage 3: out = sum over the 16 partials (fixed order)
    {
        int total = NN * EDIM;
        rgcn_reduce<<<(total + 255) / 256, 256, 0, stream>>>(partial, out);
    }
}